// ShadeWatcherGNN_51204600103258
// MI455X (gfx1250) — compile-verified
//
#include <hip/hip_runtime.h>
#include <hip/hip_bf16.h>

typedef __attribute__((ext_vector_type(2))) float v2f;
typedef __attribute__((ext_vector_type(8))) float v8f;

#define ENT_D 64
#define REL_K 32
#define NREL  16

// d_ws layout (int units)
#define WS_COUNTS 0    // 16
#define WS_START  16   // 17
#define WS_CURSOR 40   // 16
#define WS_PERM   64   // E + 16*NREL entries

__global__ void kgat_init(int* ws) {
    int i = threadIdx.x;
    if (i < NREL) { ws[WS_COUNTS + i] = 0; ws[WS_CURSOR + i] = 0; }
}

__global__ void kgat_count(const int* __restrict__ r, int E, int* __restrict__ ws) {
    __shared__ int sh[NREL];
    if (threadIdx.x < NREL) sh[threadIdx.x] = 0;
    __syncthreads();
    int i = blockIdx.x * blockDim.x + threadIdx.x;
    if (i < E) atomicAdd(&sh[r[i] & (NREL - 1)], 1);
    __syncthreads();
    if (threadIdx.x < NREL) atomicAdd(&ws[WS_COUNTS + threadIdx.x], sh[threadIdx.x]);
}

__global__ void kgat_offsets(int* ws) {
    if (threadIdx.x == 0 && blockIdx.x == 0) {
        int acc = 0;
        for (int rr = 0; rr < NREL; ++rr) {
            ws[WS_START + rr] = acc;
            int c = ws[WS_COUNTS + rr];
            acc += (c + 15) & ~15;      // pad each bucket to a 16-edge tile
        }
        ws[WS_START + NREL] = acc;
    }
}

__global__ void kgat_fill(int* ws, int cap) {
    int i = blockIdx.x * blockDim.x + threadIdx.x;
    if (i < cap) ws[WS_PERM + i] = -1;
}

__global__ void kgat_scatter(const int* __restrict__ r, int E, int* __restrict__ ws) {
    int i = blockIdx.x * blockDim.x + threadIdx.x;
    if (i < E) {
        int rr  = r[i] & (NREL - 1);
        int pos = atomicAdd(&ws[WS_CURSOR + rr], 1);
        ws[WS_PERM + ws[WS_START + rr] + pos] = i;
    }
}

// One wave32 per 16-edge tile; all edges in a tile share relation `rel`.
// A (16x64 edge embeddings) x B (64x32 W_r) accumulated via V_WMMA_F32_16X16X4_F32.
__global__ void __launch_bounds__(256)
kgat_attn(const int* __restrict__ h, const int* __restrict__ t,
          const float* __restrict__ ent, const float* __restrict__ rel_emb,
          const float* __restrict__ W, const int* __restrict__ ws,
          float* __restrict__ out) {
    const int lane  = threadIdx.x & 31;
    const int wave  = threadIdx.x >> 5;
    const int tile  = blockIdx.x * (blockDim.x >> 5) + wave;
    const int slot0 = tile * 16;
    const int total = ws[WS_START + NREL];
    if (slot0 >= total) return;                 // wave-uniform: EXEC stays all-1s

    // which relation bucket does this tile live in?
    int rel = 0;
    #pragma unroll
    for (int rr = 1; rr < NREL; ++rr)
        if (ws[WS_START + rr] <= slot0) rel = rr;

    const int* perm = ws + WS_PERM;
    const int  col  = lane & 15;
    const int  half = lane >> 4;                // 0: K={0,1}, 1: K={2,3} within chunk

    // edge feeding this lane's A-row (row index == col)
    int  e  = perm[slot0 + col];
    bool ok = (e >= 0);
    int  he = ok ? h[e] : 0;
    int  te = ok ? t[e] : 0;
    const float* hrow = ent + (long)he * ENT_D;
    const float* trow = ent + (long)te * ENT_D;
    const float* Wr   = W + (long)rel * (ENT_D * REL_K);

    v8f cH0 = {}, cH1 = {}, cT0 = {}, cT1 = {};
    #pragma unroll
    for (int kc = 0; kc < ENT_D / 4; ++kc) {
        const int k0 = kc * 4 + half * 2;
        v2f aH = { hrow[k0], hrow[k0 + 1] };                            // contiguous b64 load
        v2f aT = { trow[k0], trow[k0 + 1] };
        v2f b0 = { Wr[k0 * REL_K + col],      Wr[(k0 + 1) * REL_K + col] };
        v2f b1 = { Wr[k0 * REL_K + 16 + col], Wr[(k0 + 1) * REL_K + 16 + col] };
        cH0 = __builtin_amdgcn_wmma_f32_16x16x4_f32(false, aH, false, b0, (short)0, cH0, false, false);
        cT0 = __builtin_amdgcn_wmma_f32_16x16x4_f32(false, aT, false, b0, (short)0, cT0, false, false);
        cH1 = __builtin_amdgcn_wmma_f32_16x16x4_f32(false, aH, false, b1, (short)0, cH1, false, false);
        cT1 = __builtin_amdgcn_wmma_f32_16x16x4_f32(false, aT, false, b1, (short)0, cT1, false, false);
    }

    const float rb0 = rel_emb[rel * REL_K + col];
    const float rb1 = rel_emb[rel * REL_K + 16 + col];

    // C/D layout: VGPR i holds row (half ? i+8 : i), this lane's column = col
    #pragma unroll
    for (int i = 0; i < 8; ++i) {
        int row = half ? (i + 8) : i;
        int ei  = perm[slot0 + row];
        if (ei >= 0) {
            float a0 = cT0[i] * tanhf(cH0[i] + rb0);
            float a1 = cT1[i] * tanhf(cH1[i] + rb1);
            out[(long)ei * REL_K + col]      = a0;   // 64B-coalesced per half-wave
            out[(long)ei * REL_K + 16 + col] = a1;
        }
    }
}

extern "C" void kernel_launch(void* const* d_in, const int* in_sizes, int n_in,
                              void* d_out, int out_size, void* d_ws, size_t ws_size,
                              hipStream_t stream) {
    const int*   h       = (const int*)d_in[0];
    const int*   t       = (const int*)d_in[1];
    const int*   r       = (const int*)d_in[2];
    const float* ent     = (const float*)d_in[3];
    const float* rel_emb = (const float*)d_in[4];
    const float* W       = (const float*)d_in[5];
    float*       out     = (float*)d_out;
    int*         ws      = (int*)d_ws;

    const int E   = in_sizes[0];
    const int cap = E + 16 * NREL;              // padded permutation capacity

    kgat_init   <<<1, 32, 0, stream>>>(ws);
    kgat_count  <<<(E + 255) / 256, 256, 0, stream>>>(r, E, ws);
    kgat_offsets<<<1, 1, 0, stream>>>(ws);
    kgat_fill   <<<(cap + 255) / 256, 256, 0, stream>>>(ws, cap);
    kgat_scatter<<<(E + 255) / 256, 256, 0, stream>>>(r, E, ws);

    const int maxTiles = (E + 15) / 16 + NREL;  // upper bound incl. per-bucket padding
    const int wavesPerBlock = 8;                // 256 threads = 8 wave32 tiles/WGP pass
    const int blocks = (maxTiles + wavesPerBlock - 1) / wavesPerBlock;
    kgat_attn<<<blocks, wavesPerBlock * 32, 0, stream>>>(h, t, ent, rel_emb, W, ws, out);
}